// SO2_m_Convolution_38955353375100
// MI455X (gfx1250) — compile-verified
//
#include <hip/hip_runtime.h>

// ---------------------------------------------------------------------------
// SO2 m-convolution == one fused FP32 GEMM:
//   Out(65536 x 1280) = X(65536 x 1280) * B(1280 x 1280)
// B = [[w_pos, w_neg], [-w_neg, w_pos]] built once in phase 1, stored in a
// WMMA-fragment-packed layout Bp[k/4][n][k%4] so that each lane's B fragment
// is a single contiguous ds_load_b64 (no VGPR shuffles).
// Memory-bound: ~0.68 GB @ 23.3 TB/s ~= 29 us; FP32 WMMA keeps full precision.
// Grid order (N fastest) keeps each 655 KB X strip L2-resident across its
// 10 N-tiles; B (6.55 MB) is fully L2-resident; Out stores are non-temporal.
// ---------------------------------------------------------------------------

typedef __attribute__((ext_vector_type(2))) float v2f;
typedef __attribute__((ext_vector_type(4))) float v4f;
typedef __attribute__((ext_vector_type(8))) float v8f;

#define N_EDGES   65536
#define KDIM      1280
#define NDIM      1280
#define HALF_DIM  640

#define BM 128
#define BN 128
#define BK 32
#define LDA 36                 // padded LDS stride for A tile (floats)
#define KTILES (KDIM / BK)     // 40
#define BP_GSTRIDE (NDIM * 4)  // floats per K-group row in packed B (5120)

// ---------------------------------------------------------------------------
// Phase 1: build packed B from tp_weight / cg_pos / cg_neg.
//   o = i*128 + j, c = k*128 + l
//   w_pos[o,c] = sum_m tp[(i+2), j, (k+2), l, m] * cg_pos[i,k,m]
// Logical B (K-major):  B[c, o] = w_pos ; B[c+640, o] = -w_neg ;
//                       B[c, o+640] = w_neg ; B[c+640, o+640] = w_pos
// Packed:  Bp[(krow>>2)*5120 + ncol*4 + (krow&3)] = B[krow, ncol]
// ---------------------------------------------------------------------------
__global__ __launch_bounds__(256) void so2_build_w_kernel(
    const float* __restrict__ tp, const float* __restrict__ cgp,
    const float* __restrict__ cgn, float* __restrict__ Bp) {
  int idx = blockIdx.x * blockDim.x + threadIdx.x;
  if (idx >= HALF_DIM * HALF_DIM) return;
  int o = idx % HALF_DIM;
  int c = idx / HALF_DIM;
  int i = o >> 7, j = o & 127;
  int k = c >> 7, l = c & 127;
  // tp_weight shape (7,128,7,128,7,1); element strides:
  //   dim0: 802816, dim1: 6272, dim2: 896, dim3: 7, dim4: 1
  const float* tpp = tp + (size_t)(i + 2) * 802816 + (size_t)j * 6272 +
                     (size_t)(k + 2) * 896 + (size_t)l * 7;
  const float* cp = cgp + (i * 5 + k) * 7;
  const float* cn = cgn + (i * 5 + k) * 7;
  float sp = 0.f, sn = 0.f;
#pragma unroll
  for (int m = 0; m < 7; ++m) {
    float t = tpp[m];
    sp += t * cp[m];
    sn += t * cn[m];
  }
  size_t gLo = (size_t)(c >> 2) * BP_GSTRIDE + (c & 3);          // krow = c
  size_t gHi = gLo + (size_t)160 * BP_GSTRIDE;                   // krow = c+640
  Bp[gLo + (size_t)o * 4]              =  sp;
  Bp[gHi + (size_t)o * 4]              = -sn;
  Bp[gLo + (size_t)(o + HALF_DIM) * 4] =  sn;
  Bp[gHi + (size_t)(o + HALF_DIM) * 4] =  sp;
}

// ---------------------------------------------------------------------------
// Phase 2: FP32 WMMA GEMM with register-staged tile prefetch.
// 256 threads = 8 waves (4 along M x 2 along N); block tile 128x128x32;
// each wave: 2x4 tiles of v_wmma_f32_16x16x4_f32 (32x64 sub-tile).
// ---------------------------------------------------------------------------
__global__ __launch_bounds__(256) void so2_gemm_kernel(
    const float* __restrict__ X, const float* __restrict__ Bp,
    float* __restrict__ Out) {
  __shared__ float As[BM * LDA];   // 18432 B
  __shared__ float Bs[8 * 512];    // 16384 B : [g][n][q], q = k%4

  const int tid  = threadIdx.x;
  const int lane = tid & 31;
  const int wave = tid >> 5;
  const int wm   = wave & 3;   // 4 waves along M
  const int wn   = wave >> 2;  // 2 waves along N
  const int n0   = blockIdx.x * BN;  // N fastest -> L2 reuse of the X strip
  const int m0   = blockIdx.y * BM;

  const int half = lane >> 4;  // 0: lanes 0-15, 1: lanes 16-31
  const int l15  = lane & 15;

  // Staging index precompute (A: 1024 float4 slots, B: 1024 float4 slots).
  int rowA[4], c4A[4], gB[4], rB[4];
#pragma unroll
  for (int it = 0; it < 4; ++it) {
    int s = tid + it * 256;
    rowA[it] = s >> 3;   // 8 float4 per 32-float A row
    c4A[it]  = s & 7;
    gB[it]   = s >> 7;   // 128 float4 per packed B group
    rB[it]   = s & 127;
  }

  v8f acc[2][4];
#pragma unroll
  for (int tm = 0; tm < 2; ++tm)
#pragma unroll
    for (int tn = 0; tn < 4; ++tn) acc[tm][tn] = (v8f){};

  v4f aReg[4], bReg[4];

  // ---- prologue: fetch tile 0 and stage it ----
#pragma unroll
  for (int it = 0; it < 4; ++it) {
    aReg[it] = *(const v4f*)(X + (size_t)(m0 + rowA[it]) * KDIM + c4A[it] * 4);
    bReg[it] = *(const v4f*)(Bp + (size_t)gB[it] * BP_GSTRIDE +
                             (size_t)n0 * 4 + rB[it] * 4);
  }
#pragma unroll
  for (int it = 0; it < 4; ++it) {
    *(v4f*)(&As[rowA[it] * LDA + c4A[it] * 4]) = aReg[it];
    *(v4f*)(&Bs[gB[it] * 512 + rB[it] * 4])    = bReg[it];
  }
  __syncthreads();

  for (int t = 0; t < KTILES; ++t) {
    const int ktn = (t + 1) * BK;
    // ---- prefetch tile t+1 into registers (hidden under compute) ----
    if (t + 1 < KTILES) {
#pragma unroll
      for (int it = 0; it < 4; ++it) {
        aReg[it] = *(const v4f*)(X + (size_t)(m0 + rowA[it]) * KDIM + ktn +
                                 c4A[it] * 4);
        bReg[it] = *(const v4f*)(Bp + (size_t)(ktn >> 2) * BP_GSTRIDE +
                                 (size_t)gB[it] * BP_GSTRIDE +
                                 (size_t)n0 * 4 + rB[it] * 4);
      }
    }

    // ---- compute 8 k-steps from LDS tile t ----
#pragma unroll
    for (int g = 0; g < 8; ++g) {
      const int kk = g * 4;
      // A frag (16x4): lane -> M row; VGPR pair -> K {0,1}|{2,3} per half.
      v2f afrag[2];
#pragma unroll
      for (int tm = 0; tm < 2; ++tm) {
        int m = wm * 32 + tm * 16 + l15;
        afrag[tm] = *(const v2f*)(&As[m * LDA + kk + half * 2]);
      }
      // B frag (4x16): lane -> N col; packed layout -> single b64 per frag.
      v2f bfrag[4];
#pragma unroll
      for (int tn = 0; tn < 4; ++tn) {
        int n = wn * 64 + tn * 16 + l15;
        bfrag[tn] = *(const v2f*)(&Bs[g * 512 + n * 4 + half * 2]);
      }
#pragma unroll
      for (int tm = 0; tm < 2; ++tm)
#pragma unroll
        for (int tn = 0; tn < 4; ++tn)
          acc[tm][tn] = __builtin_amdgcn_wmma_f32_16x16x4_f32(
              /*neg_a=*/false, afrag[tm], /*neg_b=*/false, bfrag[tn],
              /*c_mod=*/(short)0, acc[tm][tn],
              /*reuse_a=*/false, /*reuse_b=*/false);
    }
    __syncthreads();  // everyone done reading LDS tile t

    if (t + 1 < KTILES) {
#pragma unroll
      for (int it = 0; it < 4; ++it) {
        *(v4f*)(&As[rowA[it] * LDA + c4A[it] * 4]) = aReg[it];
        *(v4f*)(&Bs[gB[it] * 512 + rB[it] * 4])    = bReg[it];
      }
      __syncthreads();  // tile t+1 staged
    }
  }

  // ---- store: C/D layout: VGPR r -> row (r + 8*half), lane&15 -> col ----
  // Non-temporal: Out is write-once; keep L2 for the X strips / B.
#pragma unroll
  for (int tm = 0; tm < 2; ++tm) {
#pragma unroll
    for (int tn = 0; tn < 4; ++tn) {
      int ncol = n0 + wn * 64 + tn * 16 + l15;
#pragma unroll
      for (int r = 0; r < 8; ++r) {
        int mrow = m0 + wm * 32 + tm * 16 + half * 8 + r;
        __builtin_nontemporal_store(acc[tm][tn][r],
                                    Out + (size_t)mrow * NDIM + ncol);
      }
    }
  }
}

extern "C" void kernel_launch(void* const* d_in, const int* in_sizes, int n_in,
                              void* d_out, int out_size, void* d_ws,
                              size_t ws_size, hipStream_t stream) {
  const float* x_m = (const float*)d_in[0];
  const float* tpw = (const float*)d_in[1];
  const float* cgp = (const float*)d_in[2];
  const float* cgn = (const float*)d_in[3];
  float* out = (float*)d_out;
  float* Bws = (float*)d_ws;  // packed B: 320*1280*4 floats = 6.55 MB

  // Phase 1: build packed block weight matrix into workspace.
  so2_build_w_kernel<<<(HALF_DIM * HALF_DIM + 255) / 256, 256, 0, stream>>>(
      tpw, cgp, cgn, Bws);

  // Phase 2: Out = X * B  (M=65536, K=1280, N=1280), N-tiles fastest.
  dim3 grid(NDIM / BN, N_EDGES / BM);  // (10, 512)
  so2_gemm_kernel<<<grid, 256, 0, stream>>>(x_m, Bws, out);
}